// cross_scale_trans_73615739454022
// MI455X (gfx1250) — compile-verified
//
#include <hip/hip_runtime.h>
#include <hip/hip_bf16.h>

typedef float v2f __attribute__((ext_vector_type(2)));
typedef float v8f __attribute__((ext_vector_type(8)));

__device__ __forceinline__ v8f wmma_f32_16x16x4(v2f a, v2f b, v8f c) {
    // D = A(16x4 f32) * B(4x16 f32) + C(16x16 f32), wave32
    return __builtin_amdgcn_wmma_f32_16x16x4_f32(false, a, false, b, (short)0, c,
                                                 false, false);
}

// out[16 x 16*NT] = A[16 x 4*KSTEPS] * B[4*KSTEPS x 16*NT] + bias (fp32 WMMA)
// A layout per ISA 7.12.2: lane m=L%16.., vgpr0: K = 4k + 2*hi, vgpr1: K+1
template <int NT, int KSTEPS>
__device__ __forceinline__ void wmma_tile_gemm(const float* A, int lda,
                                               const float* __restrict__ B, int ldb,
                                               const float* __restrict__ bias,
                                               float* outp, int ldo, int lane) {
    const int m = lane & 15;
    const int hi = lane >> 4;
    v8f acc[NT];
#pragma unroll
    for (int j = 0; j < NT; ++j) {
        float bv = bias[16 * j + m];
        v8f t = {bv, bv, bv, bv, bv, bv, bv, bv};
        acc[j] = t;
    }
#pragma unroll
    for (int k0 = 0; k0 < KSTEPS; ++k0) {
        const int kk = 4 * k0 + 2 * hi;
        v2f a;
        a.x = A[m * lda + kk];
        a.y = A[m * lda + kk + 1];
#pragma unroll
        for (int j = 0; j < NT; ++j) {
            const int col = 16 * j + m;
            v2f b;
            b.x = B[kk * ldb + col];
            b.y = B[(kk + 1) * ldb + col];
            acc[j] = wmma_f32_16x16x4(a, b, acc[j]);
        }
    }
    // C/D layout: vgpr r holds row (r + 8*hi), col = 16*j + m
#pragma unroll
    for (int j = 0; j < NT; ++j)
#pragma unroll
        for (int r = 0; r < 8; ++r)
            outp[(r + 8 * hi) * ldo + 16 * j + m] = acc[j][r];
}

// One wave32 per 16-voxel tile: GEMM chain + groupnorm + softmax + scatter +
// sample-index generation.
template <int C, int P>
__global__ __launch_bounds__(32) void level_compute_kernel(
    const float* __restrict__ feat, const int* __restrict__ idx,
    const float* __restrict__ in_W, const float* __restrict__ in_b,
    const float* __restrict__ gn_g, const float* __restrict__ gn_b,
    const float* __restrict__ q_W, const float* __restrict__ q_b,
    const float* __restrict__ v_W, const float* __restrict__ v_b,
    const float* __restrict__ off_W, const float* __restrict__ off_b,
    const float* __restrict__ aw_W, const float* __restrict__ aw_b,
    float* __restrict__ vol, int* __restrict__ sidx, float* __restrict__ saw,
    int D, int H, int W) {
    constexpr int NP = 4 * P;     // heads * points (32 or 16)
    constexpr int NOFF = NP * 3;  // offset columns (96 or 48)
    constexpr int S = 64 / C;     // group size (groups == C)

    __shared__ float s_feat[16][C];
    __shared__ float s_src[16][64];
    __shared__ float s_q[16][C];
    __shared__ float s_v[16][C];
    __shared__ float s_off[16][NOFF];
    __shared__ float s_aw[16][NP];

    const int lane = threadIdx.x;
    const int base = blockIdx.x * 16;

    // ---- load feature tile [16 x C] ----
    for (int t = lane; t < 16 * C; t += 32)
        s_feat[t / C][t % C] = feat[(size_t)(base + t / C) * C + (t % C)];
    __syncthreads();

    // ---- src = feat @ in_W + in_b   [16 x 64] ----
    wmma_tile_gemm<4, C / 4>(&s_feat[0][0], C, in_W, 64, in_b, &s_src[0][0], 64, lane);
    __syncthreads();

    // ---- GroupNorm(groups=C) over D_MODEL=64, biased variance ----
    for (int t = lane; t < 16 * C; t += 32) {
        const int row = t / C, g = t % C;
        float mu = 0.f;
#pragma unroll
        for (int s = 0; s < S; ++s) mu += s_src[row][g * S + s];
        mu *= (1.0f / S);
        float var = 0.f;
#pragma unroll
        for (int s = 0; s < S; ++s) {
            float d = s_src[row][g * S + s] - mu;
            var += d * d;
        }
        var *= (1.0f / S);
        const float inv = rsqrtf(var + 1e-5f);
#pragma unroll
        for (int s = 0; s < S; ++s) {
            const int ch = g * S + s;
            s_src[row][ch] = (s_src[row][ch] - mu) * inv * gn_g[ch] + gn_b[ch];
        }
    }
    __syncthreads();

    // ---- query / value = src @ {q_W, v_W} + bias   [16 x C] ----
    wmma_tile_gemm<C / 16, 16>(&s_src[0][0], 64, q_W, C, q_b, &s_q[0][0], C, lane);
    wmma_tile_gemm<C / 16, 16>(&s_src[0][0], 64, v_W, C, v_b, &s_v[0][0], C, lane);
    __syncthreads();

    // ---- off = query @ off_W + off_b; aw_logits = query @ aw_W + aw_b ----
    wmma_tile_gemm<NOFF / 16, C / 4>(&s_q[0][0], C, off_W, NOFF, off_b, &s_off[0][0],
                                     NOFF, lane);
    wmma_tile_gemm<NP / 16, C / 4>(&s_q[0][0], C, aw_W, NP, aw_b, &s_aw[0][0], NP, lane);
    __syncthreads();

    // ---- softmax over flat NP axis, one row per lane ----
    if (lane < 16) {
        const int row = lane;
        float mx = s_aw[row][0];
#pragma unroll
        for (int j = 1; j < NP; ++j) mx = fmaxf(mx, s_aw[row][j]);
        float sum = 0.f;
#pragma unroll
        for (int j = 0; j < NP; ++j) {
            const float e = expf(s_aw[row][j] - mx);
            s_aw[row][j] = e;
            sum += e;
        }
        const float r = 1.0f / sum;
#pragma unroll
        for (int j = 0; j < NP; ++j) s_aw[row][j] *= r;
    }
    __syncthreads();

    // ---- scatter value into dense volume [D*H*W][C] ----
    for (int t = lane; t < 16 * C; t += 32) {
        const int row = t / C, ch = t % C;
        const int vz = idx[(size_t)(base + row) * 3 + 0];
        const int vy = idx[(size_t)(base + row) * 3 + 1];
        const int vx = idx[(size_t)(base + row) * 3 + 2];
        const size_t flat = (size_t)((vz * H + vy) * W + vx);
        vol[flat * C + ch] = s_v[row][ch];
    }

    // ---- sample indices + weights ----
    const float Df = (float)D, Hf = (float)H, Wf = (float)W;
    for (int t = lane; t < 16 * NP; t += 32) {
        const int row = t / NP, j = t % NP;
        const int vz = idx[(size_t)(base + row) * 3 + 0];
        const int vy = idx[(size_t)(base + row) * 3 + 1];
        const int vx = idx[(size_t)(base + row) * 3 + 2];
        const float o0 = s_off[row][j * 3 + 0];
        const float o1 = s_off[row][j * 3 + 1];
        const float o2 = s_off[row][j * 3 + 2];
        // grid_sample convention of the reference (comp0 norm'd by D, sampled vs W)
        const float loc0 = o0 / Df + (float)vz / Df;
        const float loc1 = o1 / Hf + (float)vy / Hf;
        const float loc2 = o2 / Wf + (float)vx / Wf;
        const float g0 = 2.0f * loc0 - 1.0f;
        const float g1 = 2.0f * loc1 - 1.0f;
        const float g2 = 2.0f * loc2 - 1.0f;
        const int ix = (int)rintf((g0 + 1.0f) * (Wf * 0.5f) - 0.5f);
        const int iy = (int)rintf((g1 + 1.0f) * (Hf * 0.5f) - 0.5f);
        const int iz = (int)rintf((g2 + 1.0f) * (Df * 0.5f) - 0.5f);
        const bool valid =
            (ix >= 0) && (ix < W) && (iy >= 0) && (iy < H) && (iz >= 0) && (iz < D);
        const int flat = (iz * H + iy) * W + ix;
        sidx[(size_t)(base + row) * NP + j] = valid ? flat : -1;
        saw[(size_t)(base + row) * NP + j] = s_aw[row][j];
    }
}

// Gather: out[v][ch] = sum_s aw[v][s] * vol[flat(v,s)][ch]
template <int C, int P>
__global__ __launch_bounds__(256) void level_gather_kernel(
    const float* __restrict__ vol, const int* __restrict__ sidx,
    const float* __restrict__ saw, float* __restrict__ out, int n) {
    constexpr int NP = 4 * P;
    constexpr int VPB = 256 / C;
    const int t = threadIdx.x;
    const int v = blockIdx.x * VPB + t / C;
    const int ch = t % C;
    if (v >= n) return;
    const int* si = sidx + (size_t)v * NP;
    const float* sw = saw + (size_t)v * NP;
    float acc = 0.f;
#pragma unroll
    for (int s = 0; s < NP; ++s) {
        const int f = si[s];
        if (f >= 0) acc += sw[s] * vol[(size_t)f * C + ch];
    }
    out[(size_t)v * C + ch] = acc;
}

// ---------------- host side ----------------

static const int NVOXL[4] = {100000, 50000, 20000, 8000};
static const int DDL[4] = {11, 5, 5, 3};
static const int HHL[4] = {400, 200, 100, 50};
static const int WWL[4] = {352, 176, 88, 44};
static const int CCL[4] = {16, 32, 64, 64};
static const int PPL[4] = {8, 8, 4, 4};

template <int C, int P>
static void run_level(const float* feat, const int* idx, const float* const* prm,
                      float* vol, int* sidx, float* saw, float* out, int n, int D,
                      int H, int W, hipStream_t stream) {
    dim3 g1(n / 16), b1(32);
    level_compute_kernel<C, P><<<g1, b1, 0, stream>>>(
        feat, idx, prm[0], prm[1], prm[2], prm[3], prm[4], prm[5], prm[6], prm[7],
        prm[8], prm[9], prm[10], prm[11], vol, sidx, saw, D, H, W);
    dim3 g2(n / (256 / C)), b2(256);
    level_gather_kernel<C, P><<<g2, b2, 0, stream>>>(vol, sidx, saw, out, n);
}

extern "C" void kernel_launch(void* const* d_in, const int* in_sizes, int n_in,
                              void* d_out, int out_size, void* d_ws, size_t ws_size,
                              hipStream_t stream) {
    (void)in_sizes; (void)n_in; (void)out_size; (void)ws_size;

    float* ws = (float*)d_ws;
    float* out = (float*)d_out;

    // workspace layout (element offsets)
    size_t volOff[4], sidxOff[4], sawOff[4];
    size_t cur = 0;
    for (int l = 0; l < 4; ++l) {
        volOff[l] = cur;
        cur += (size_t)DDL[l] * HHL[l] * WWL[l] * CCL[l];
    }
    const size_t volTotal = cur;
    for (int l = 0; l < 4; ++l) {
        sidxOff[l] = cur;
        cur += (size_t)NVOXL[l] * 4 * PPL[l];
        sawOff[l] = cur;
        cur += (size_t)NVOXL[l] * 4 * PPL[l];
    }

    // zero the dense volumes (scatter only touches occupied voxels)
    hipMemsetAsync(ws, 0, volTotal * sizeof(float), stream);

    // inputs: feats0, idx0, feats1, idx1, ..., then 12 params per level
    size_t outOff = 0;
    for (int l = 0; l < 4; ++l) {
        const float* feat = (const float*)d_in[2 * l];
        const int* idx = (const int*)d_in[2 * l + 1];
        const float* prm[12];
        for (int k = 0; k < 12; ++k) prm[k] = (const float*)d_in[8 + 12 * l + k];

        float* vol = ws + volOff[l];
        int* sidx = (int*)(ws + sidxOff[l]);
        float* saw = ws + sawOff[l];
        float* outl = out + outOff;

        switch (l) {
            case 0:
                run_level<16, 8>(feat, idx, prm, vol, sidx, saw, outl, NVOXL[l],
                                 DDL[l], HHL[l], WWL[l], stream);
                break;
            case 1:
                run_level<32, 8>(feat, idx, prm, vol, sidx, saw, outl, NVOXL[l],
                                 DDL[l], HHL[l], WWL[l], stream);
                break;
            case 2:
                run_level<64, 4>(feat, idx, prm, vol, sidx, saw, outl, NVOXL[l],
                                 DDL[l], HHL[l], WWL[l], stream);
                break;
            default:
                run_level<64, 4>(feat, idx, prm, vol, sidx, saw, outl, NVOXL[l],
                                 DDL[l], HHL[l], WWL[l], stream);
                break;
        }
        outOff += (size_t)NVOXL[l] * CCL[l];
    }
}